// GCN_42271068127247
// MI455X (gfx1250) — compile-verified
//
#include <hip/hip_runtime.h>
#include <hip/hip_bf16.h>

typedef __attribute__((ext_vector_type(2))) float v2f;
typedef __attribute__((ext_vector_type(8))) float v8f;

#define DEG 16
#define D 64
#define TWO_D 128
#define ZS 132                 // padded LDS row stride in floats; 132*4B = 33*16B -> rows stay 16B aligned
#define NODES_PER_BLOCK 32

// Fused GCN layer: pool = segment_sum(ft[nbr]); z = [ft+pool, ft*pool];
// out = relu(z @ W^T)  with W [64,128] row-major.
__global__ __launch_bounds__(256) void gcn_layer_kernel(
    const float* __restrict__ ft, const float* __restrict__ W,
    const int* __restrict__ nbr, float* __restrict__ out, int n_nodes)
{
    __shared__ float Wlds[D * ZS];                 // W[o][k] at Wlds[o*ZS+k]
    __shared__ float zlds[NODES_PER_BLOCK * ZS];   // z tile, 32 x 128 (padded)
    __shared__ int   nb[NODES_PER_BLOCK * DEG];    // staged neighbor ids

    const int t = threadIdx.x;
    const int blockNode0 = blockIdx.x * NODES_PER_BLOCK;

    // Stage W (64x128 floats = 2048 float4) into LDS with b128 ops.
    {
        const float4* W4 = (const float4*)W;
        #pragma unroll
        for (int i = t; i < (D * TWO_D) / 4; i += 256) {
            int o = i >> 5;           // row (128 floats = 32 float4 per row)
            int q = i & 31;           // float4 chunk within row
            *(float4*)&Wlds[o * ZS + 4 * q] = W4[i];
        }
    }
    // Stage neighbor indices (32*16 = 512 ints).
    for (int i = t; i < NODES_PER_BLOCK * DEG; i += 256) {
        int node = blockNode0 + (i >> 4);
        nb[i] = (node < n_nodes) ? nbr[node * DEG + (i & 15)] : 0;
    }
    __syncthreads();

    // Gather + pool + concat into LDS z tile, float4 granularity.
    // 32 nodes x 16 float4-chunks = 512 slots; 2 slots per thread.
    // 16 consecutive lanes cover one neighbor row -> coalesced b128 loads.
    #pragma unroll
    for (int slot = t; slot < NODES_PER_BLOCK * (D / 4); slot += 256) {
        int ln = slot >> 4;             // local node 0..31
        int q  = slot & 15;             // float4 chunk: dims 4q..4q+3
        int node = blockNode0 + ln;
        float4 f  = make_float4(0.f, 0.f, 0.f, 0.f);
        float4 pl = make_float4(0.f, 0.f, 0.f, 0.f);
        if (node < n_nodes) {
            f = *(const float4*)&ft[node * D + 4 * q];
            #pragma unroll
            for (int k = 0; k < DEG; ++k) {
                float4 v = *(const float4*)&ft[nb[ln * DEG + k] * D + 4 * q];
                pl.x += v.x; pl.y += v.y; pl.z += v.z; pl.w += v.w;
            }
        }
        *(float4*)&zlds[ln * ZS + 4 * q] =
            make_float4(f.x + pl.x, f.y + pl.y, f.z + pl.z, f.w + pl.w);
        *(float4*)&zlds[ln * ZS + D + 4 * q] =
            make_float4(f.x * pl.x, f.y * pl.y, f.z * pl.z, f.w * pl.w);
    }
    __syncthreads();

    // 8 waves: wave w handles node tile (w&1)*16, output tile (w>>1)*16.
    const int wave = t >> 5;
    const int lane = t & 31;
    const int m16  = lane & 15;
    const int hi   = lane >> 4;          // lane group (adds 2 to K in A/B frags)
    const int nodeTile = (wave & 1) * 16;
    const int outTile  = (wave >> 1) * 16;

    // A fragment: A[m][k] = z[nodeTile+m][k],  m = lane%16, k = k0 + 2*hi + v
    // B fragment: B[k][o] = W[outTile+o][k],   o = lane%16, k = k0 + 2*hi + v
    const float* Az = &zlds[(nodeTile + m16) * ZS + 2 * hi];
    const float* Bw = &Wlds[(outTile  + m16) * ZS + 2 * hi];

    v8f c = {};
    #pragma unroll
    for (int k0 = 0; k0 < TWO_D; k0 += 4) {
        v2f a = *(const v2f*)(Az + k0);   // 8B-aligned LDS loads
        v2f b = *(const v2f*)(Bw + k0);
        // D = A(16x4,f32) * B(4x16,f32) + C(16x16,f32)
        c = __builtin_amdgcn_wmma_f32_16x16x4_f32(
                /*neg_a=*/false, a, /*neg_b=*/false, b,
                /*c_mod=*/(short)0, c, /*reuse_a=*/false, /*reuse_b=*/false);
    }

    // C/D layout: VGPR r -> M = r + 8*hi, N = lane%16.  ReLU + predicated store.
    #pragma unroll
    for (int r = 0; r < 8; ++r) {
        int row = blockNode0 + nodeTile + r + 8 * hi;
        if (row < n_nodes) {
            float x = c[r];
            out[row * D + outTile + m16] = x > 0.f ? x : 0.f;
        }
    }
}

// L2 normalize each 64-wide row; one wave32 per node, 2 elements per lane.
__global__ __launch_bounds__(256) void l2norm_kernel(
    const float* __restrict__ in, float* __restrict__ out, int n_nodes)
{
    int node = blockIdx.x * 8 + (threadIdx.x >> 5);
    int lane = threadIdx.x & 31;
    if (node >= n_nodes) return;
    float x0 = in[node * D + lane];
    float x1 = in[node * D + 32 + lane];
    float s = x0 * x0 + x1 * x1;
    #pragma unroll
    for (int off = 16; off; off >>= 1)
        s += __shfl_xor(s, off, 32);
    float inv = 1.0f / fmaxf(sqrtf(s), 1e-12f);
    out[node * D + lane]      = x0 * inv;
    out[node * D + 32 + lane] = x1 * inv;
}

extern "C" void kernel_launch(void* const* d_in, const int* in_sizes, int n_in,
                              void* d_out, int out_size, void* d_ws, size_t ws_size,
                              hipStream_t stream) {
    const float* ft0 = (const float*)d_in[0];   // [N,64]
    const float* W1  = (const float*)d_in[1];   // [64,128]
    const float* W2  = (const float*)d_in[2];   // [64,128]
    const int*   nbr = (const int*)d_in[3];     // [N*16]
    // d_in[4] (tgt) is implicit: sorted, DEG edges per node.

    const int n = in_sizes[0] / D;              // 50000

    float* ft1 = (float*)d_ws;                  // [N,64]
    float* ft2 = ft1 + (size_t)n * D;           // [N,64]

    const int layer_blocks = (n + NODES_PER_BLOCK - 1) / NODES_PER_BLOCK;
    gcn_layer_kernel<<<layer_blocks, 256, 0, stream>>>(ft0, W1, nbr, ft1, n);
    gcn_layer_kernel<<<layer_blocks, 256, 0, stream>>>(ft1, W2, nbr, ft2, n);

    const int norm_blocks = (n + 7) / 8;
    l2norm_kernel<<<norm_blocks, 256, 0, stream>>>(ft2, (float*)d_out, n);
}